// HANLayer_89850715832642
// MI455X (gfx1250) — compile-verified
//
#include <hip/hip_runtime.h>
#include <hip/hip_bf16.h>

#ifndef __has_builtin
#define __has_builtin(x) 0
#endif
#if __has_builtin(__builtin_amdgcn_wmma_f32_16x16x4_f32)
#define USE_WMMA_F32 1
#else
#define USE_WMMA_F32 0
#endif

typedef float v2f __attribute__((ext_vector_type(2)));
typedef float v8f __attribute__((ext_vector_type(8)));

#define NEG_SLOPE 0.2f

// ---------------------------------------------------------------------------
// Z[N,128] = X[N,128] @ W[128,128], fp32 WMMA (V_WMMA_F32_16X16X4_F32).
// Block = 256 threads = 8 waves; wave w computes the 16x16 tile at columns
// [16w, 16w+16) for the 16-row slab row0..row0+15.
// A frag (16x4 f32): lane<16 holds row m=lane, K={k,k+1}; lane>=16 same row,
// K={k+2,k+3}.  B frag (4x16): lane<16 rows k,k+1; lane>=16 rows k+2,k+3.
// C/D: VGPR r, lanes 0-15 -> (row0+r, col), lanes 16-31 -> (row0+8+r, col).
// ---------------------------------------------------------------------------
__global__ void __launch_bounds__(256) proj_gemm(const float* __restrict__ X,
                                                 const float* __restrict__ W,
                                                 float* __restrict__ Z, int N) {
  const int lane = threadIdx.x & 31;
  const int wave = threadIdx.x >> 5;
  const int row0 = blockIdx.x * 16;
  const int col0 = wave * 16;
  const int m    = lane & 15;
  const int koff = (lane >> 4) * 2;
  int arow = row0 + m; if (arow >= N) arow = N - 1;   // clamp (keeps EXEC full)
  v8f acc = {0.f, 0.f, 0.f, 0.f, 0.f, 0.f, 0.f, 0.f};
#if USE_WMMA_F32
  for (int k = 0; k < 128; k += 4) {
    v2f a, b;
    a.x = X[arow * 128 + k + koff + 0];
    a.y = X[arow * 128 + k + koff + 1];
    b.x = W[(k + koff + 0) * 128 + col0 + m];
    b.y = W[(k + koff + 1) * 128 + col0 + m];
    acc = __builtin_amdgcn_wmma_f32_16x16x4_f32(false, a, false, b, (short)0,
                                                acc, false, false);
  }
#else
  { // scalar fallback (keeps build green if builtin signature differs)
    const int lr = (lane >> 4) ? 8 : 0;
    for (int r = 0; r < 8; ++r) {
      int rr = row0 + lr + r; if (rr >= N) rr = N - 1;
      float s = 0.f;
      for (int k = 0; k < 128; ++k) s = fmaf(X[rr * 128 + k], W[k * 128 + col0 + m], s);
      acc[r] = s;
    }
  }
#endif
  const int rbase = row0 + ((lane >> 4) ? 8 : 0);
  const int colw  = col0 + m;
#pragma unroll
  for (int r = 0; r < 8; ++r) {
    const int row = rbase + r;
    if (row < N) Z[row * 128 + colw] = acc[r];
  }
}

// out[n*8+h] = sum_d Z[n,16h+d] * a[h,d]      (el / er from a stored z)
__global__ void head_dot(const float* __restrict__ Z, const float* __restrict__ a,
                         float* __restrict__ out, int N) {
  const int idx = blockIdx.x * blockDim.x + threadIdx.x;
  if (idx >= N * 8) return;
  const int n = idx >> 3, h = idx & 7;
  const float* z = Z + n * 128 + h * 16;
  const float* av = a + h * 16;
  float s = 0.f;
#pragma unroll
  for (int d = 0; d < 16; ++d) s = fmaf(z[d], av[d], s);
  out[idx] = s;
}

// er[n,h] = (X[n,:] @ W)[16h:16h+16] . ar[h,:]  computed as X @ (W @ ar_h),
// with wv = W@ar folded once per block into LDS (avoids storing z for
// dst-only projections).
__global__ void __launch_bounds__(256) er_direct(const float* __restrict__ X,
                                                 const float* __restrict__ W,
                                                 const float* __restrict__ ar,
                                                 float* __restrict__ out, int N) {
  __shared__ float wv[128 * 8];
  const int tid = threadIdx.x;
  for (int i = tid; i < 128 * 8; i += 256) {
    const int k = i >> 3, h = i & 7;
    float s = 0.f;
#pragma unroll
    for (int d = 0; d < 16; ++d) s = fmaf(W[k * 128 + h * 16 + d], ar[h * 16 + d], s);
    wv[i] = s;
  }
  __syncthreads();
  const int n = blockIdx.x * 32 + (tid >> 3);
  const int h = tid & 7;
  if (n >= N) return;
  const float* x = X + n * 128;
  float s = 0.f;
  for (int k = 0; k < 128; ++k) s = fmaf(x[k], wv[k * 8 + h], s);
  out[n * 8 + h] = s;
}

__global__ void fill_u32(unsigned* __restrict__ p, unsigned v, int n) {
  const int i = blockIdx.x * blockDim.x + threadIdx.x;
  if (i < n) p[i] = v;
}

// o[n*128+d] = b[d]   (segment-sum accumulators start at the bias)
__global__ void init_bias(float* __restrict__ o, const float* __restrict__ b, int n128) {
  const int i = blockIdx.x * blockDim.x + threadIdx.x;
  if (i < n128) o[i] = b[i & 127];
}

__device__ __forceinline__ unsigned enc_f(float f) {
  unsigned b = __float_as_uint(f);
  return (b & 0x80000000u) ? ~b : (b | 0x80000000u);   // order-preserving
}
__device__ __forceinline__ float dec_f(unsigned k) {
  unsigned b = (k & 0x80000000u) ? (k & 0x7FFFFFFFu) : ~k;
  return __uint_as_float(b);
}
__device__ __forceinline__ float lrelu(float v) { return v > 0.f ? v : NEG_SLOPE * v; }

__global__ void edge_max(const int* __restrict__ src, const int* __restrict__ dst,
                         const float* __restrict__ el, const float* __restrict__ er,
                         unsigned* __restrict__ emax, int E) {
  const int idx = blockIdx.x * blockDim.x + threadIdx.x;
  if (idx >= E * 8) return;
  const int e = idx >> 3, h = idx & 7;
  const int s = src[e], d = dst[e];
  const float v = lrelu(el[s * 8 + h] + er[d * 8 + h]);
  atomicMax(&emax[d * 8 + h], enc_f(v));
}

__global__ void emax_decode(unsigned* __restrict__ eu, float* __restrict__ ef, int n) {
  const int i = blockIdx.x * blockDim.x + threadIdx.x;
  if (i >= n) return;
  float f = dec_f(eu[i]);
  if (!isfinite(f)) f = 0.f;              // nodes with no in-edges: -inf -> 0
  ef[i] = f;
}

__global__ void edge_sum(const int* __restrict__ src, const int* __restrict__ dst,
                         const float* __restrict__ el, const float* __restrict__ er,
                         const float* __restrict__ emax, float* __restrict__ den, int E) {
  const int idx = blockIdx.x * blockDim.x + threadIdx.x;
  if (idx >= E * 8) return;
  const int e = idx >> 3, h = idx & 7;
  const int s = src[e], d = dst[e];
  const float v = lrelu(el[s * 8 + h] + er[d * 8 + h]);
  atomicAdd(&den[d * 8 + h], expf(v - emax[d * 8 + h]));
}

// 32 lanes per edge; lane l owns dims [4l,4l+4) (head h = l/4): float4 gather
// of z_src + 4 f32 scatter atomics. attn recomputed (saves 144MB of per-edge
// storage; all scalar arrays are L2-resident on MI455X).
__global__ void __launch_bounds__(256) edge_scatter(
    const int* __restrict__ src, const int* __restrict__ dst,
    const float* __restrict__ el, const float* __restrict__ er,
    const float* __restrict__ emax, const float* __restrict__ den,
    const float* __restrict__ Z, float* __restrict__ out, int E) {
  const int gid = blockIdx.x * 256 + threadIdx.x;
  const int e = gid >> 5;
  if (e >= E) return;
  const int lane = gid & 31;
  const int h = lane >> 2;
  const int s = src[e], d = dst[e];
  const float v = lrelu(el[s * 8 + h] + er[d * 8 + h]);
  const float ex = expf(v - emax[d * 8 + h]);
  const float attn = ex / fmaxf(den[d * 8 + h], 1e-9f);
  const float4 z4 = *reinterpret_cast<const float4*>(Z + s * 128 + lane * 4);
  float* o = out + d * 128 + lane * 4;
  atomicAdd(o + 0, attn * z4.x);
  atomicAdd(o + 1, attn * z4.y);
  atomicAdd(o + 2, attn * z4.z);
  atomicAdd(o + 3, attn * z4.w);
}

// s[n*R+r] = tanh(o[n,:]@W1 + b1) @ W2 + b2 ; one wave per node.
__global__ void __launch_bounds__(256) semantic_score(
    const float* __restrict__ o, const float* __restrict__ W1,
    const float* __restrict__ b1, const float* __restrict__ W2,
    const float* __restrict__ b2, float* __restrict__ sout, int N, int R, int r) {
  __shared__ float xrow[8][128];
  const int tid = threadIdx.x;
  const int wave = tid >> 5, lane = tid & 31;
  const int node = blockIdx.x * 8 + wave;
  if (node < N)
    for (int i = lane; i < 128; i += 32) xrow[wave][i] = o[node * 128 + i];
  __syncthreads();
  if (node >= N) return;
  float partial = 0.f;
#pragma unroll
  for (int jj = 0; jj < 4; ++jj) {
    const int j = lane + jj * 32;
    float acc = b1[j];
    for (int k = 0; k < 128; ++k) acc = fmaf(xrow[wave][k], W1[k * 128 + j], acc);
    partial += tanhf(acc) * W2[j];
  }
  for (int off = 16; off > 0; off >>= 1) partial += __shfl_down(partial, off, 32);
  if (lane == 0) sout[node * R + r] = partial + b2[0];
}

// softmax over relations + weighted sum -> final output
__global__ void semantic_combine(const float* __restrict__ o0,
                                 const float* __restrict__ o1,
                                 const float* __restrict__ o2,
                                 const float* __restrict__ s,
                                 float* __restrict__ out, int N, int R) {
  const int idx = blockIdx.x * blockDim.x + threadIdx.x;
  if (idx >= N * 128) return;
  const int n = idx >> 7;
  float sc[3];
  float m = -3.0e38f;
  for (int r = 0; r < R; ++r) { sc[r] = s[n * R + r]; m = fmaxf(m, sc[r]); }
  float dsum = 0.f;
  for (int r = 0; r < R; ++r) { sc[r] = expf(sc[r] - m); dsum += sc[r]; }
  const float* os[3] = {o0, o1, o2};
  float acc = 0.f;
  for (int r = 0; r < R; ++r) acc += (sc[r] / dsum) * os[r][idx];
  out[idx] = acc;
}

// ---------------------------------------------------------------------------
extern "C" void kernel_launch(void* const* d_in, const int* in_sizes, int n_in,
                              void* d_out, int out_size, void* d_ws, size_t ws_size,
                              hipStream_t stream) {
  const float* h_drug = (const float*)d_in[0];
  const float* h_prot = (const float*)d_in[1];
  const float* h_se   = (const float*)d_in[2];
  const float *W_dd = (const float*)d_in[3],  *al_dd = (const float*)d_in[4],
              *ar_dd = (const float*)d_in[5], *b_dd = (const float*)d_in[6];
  const float *W_dp = (const float*)d_in[7],  *al_dp = (const float*)d_in[8],
              *ar_dp = (const float*)d_in[9], *b_dp = (const float*)d_in[10];
  const float *W_pd = (const float*)d_in[11], *al_pd = (const float*)d_in[12],
              *ar_pd = (const float*)d_in[13],*b_pd = (const float*)d_in[14];
  const float *W_pp = (const float*)d_in[15], *al_pp = (const float*)d_in[16],
              *ar_pp = (const float*)d_in[17],*b_pp = (const float*)d_in[18];
  const float *W_sd = (const float*)d_in[19], *al_sd = (const float*)d_in[20],
              *ar_sd = (const float*)d_in[21],*b_sd = (const float*)d_in[22];
  const float *W1 = (const float*)d_in[23], *b1 = (const float*)d_in[24];
  const float *W2 = (const float*)d_in[25], *b2 = (const float*)d_in[26];
  const int *src_dd = (const int*)d_in[27], *dst_dd = (const int*)d_in[28];
  const int *src_dp = (const int*)d_in[29], *dst_dp = (const int*)d_in[30];
  const int *src_pd = (const int*)d_in[31], *dst_pd = (const int*)d_in[32];
  const int *src_pp = (const int*)d_in[33], *dst_pp = (const int*)d_in[34];
  const int *src_sd = (const int*)d_in[35], *dst_sd = (const int*)d_in[36];

  const int ND = in_sizes[0] / 128, NP = in_sizes[1] / 128, NS = in_sizes[2] / 128;
  const int E_dd = in_sizes[27], E_dp = in_sizes[29], E_pd = in_sizes[31],
            E_pp = in_sizes[33], E_sd = in_sizes[35];

  // -------- workspace carve (floats) --------
  float* w = (float*)d_ws;
  size_t off = 0;
  auto alloc = [&](size_t n) { float* p = w + off; off += n; return p; };
  float* z_dd   = alloc((size_t)ND * 128);
  float* z_dp_d = alloc((size_t)ND * 128);
  float* z_pd_p = alloc((size_t)NP * 128);
  float* z_pp   = alloc((size_t)NP * 128);
  float* z_sd_s = alloc((size_t)NS * 128);
  float* el_dd = alloc((size_t)ND * 8); float* er_dd = alloc((size_t)ND * 8);
  float* el_dp = alloc((size_t)ND * 8); float* er_dp = alloc((size_t)NP * 8);
  float* el_pd = alloc((size_t)NP * 8); float* er_pd = alloc((size_t)ND * 8);
  float* el_pp = alloc((size_t)NP * 8); float* er_pp = alloc((size_t)NP * 8);
  float* el_sd = alloc((size_t)NS * 8); float* er_sd = alloc((size_t)ND * 8);
  float* emax_dd = alloc((size_t)ND * 8); float* emax_dp = alloc((size_t)NP * 8);
  float* emax_pd = alloc((size_t)ND * 8); float* emax_pp = alloc((size_t)NP * 8);
  float* emax_sd = alloc((size_t)ND * 8);
  float* den_dd = alloc((size_t)ND * 8); float* den_dp = alloc((size_t)NP * 8);
  float* den_pd = alloc((size_t)ND * 8); float* den_pp = alloc((size_t)NP * 8);
  float* den_sd = alloc((size_t)ND * 8);
  float* o_dd = alloc((size_t)ND * 128); float* o_pd = alloc((size_t)ND * 128);
  float* o_sd = alloc((size_t)ND * 128);
  float* o_dp = alloc((size_t)NP * 128); float* o_pp = alloc((size_t)NP * 128);
  float* s_drug = alloc((size_t)ND * 3); float* s_prot = alloc((size_t)NP * 2);

  const int T = 256;
  auto blk = [&](long long n) { return (unsigned)((n + T - 1) / T); };

  // 1) projections (WMMA GEMMs)
  proj_gemm<<<(ND + 15) / 16, T, 0, stream>>>(h_drug, W_dd, z_dd,   ND);
  proj_gemm<<<(ND + 15) / 16, T, 0, stream>>>(h_drug, W_dp, z_dp_d, ND);
  proj_gemm<<<(NP + 15) / 16, T, 0, stream>>>(h_prot, W_pd, z_pd_p, NP);
  proj_gemm<<<(NP + 15) / 16, T, 0, stream>>>(h_prot, W_pp, z_pp,   NP);
  proj_gemm<<<(NS + 15) / 16, T, 0, stream>>>(h_se,   W_sd, z_sd_s, NS);

  // 2) attention logits from stored z
  head_dot<<<blk((long long)ND * 8), T, 0, stream>>>(z_dd,   al_dd, el_dd, ND);
  head_dot<<<blk((long long)ND * 8), T, 0, stream>>>(z_dd,   ar_dd, er_dd, ND);
  head_dot<<<blk((long long)ND * 8), T, 0, stream>>>(z_dp_d, al_dp, el_dp, ND);
  head_dot<<<blk((long long)NP * 8), T, 0, stream>>>(z_pd_p, al_pd, el_pd, NP);
  head_dot<<<blk((long long)NP * 8), T, 0, stream>>>(z_pp,   al_pp, el_pp, NP);
  head_dot<<<blk((long long)NP * 8), T, 0, stream>>>(z_pp,   ar_pp, er_pp, NP);
  head_dot<<<blk((long long)NS * 8), T, 0, stream>>>(z_sd_s, al_sd, el_sd, NS);
  // dst-only projections folded (no z stored)
  er_direct<<<(ND + 31) / 32, T, 0, stream>>>(h_drug, W_pd, ar_pd, er_pd, ND);
  er_direct<<<(ND + 31) / 32, T, 0, stream>>>(h_drug, W_sd, ar_sd, er_sd, ND);
  er_direct<<<(NP + 31) / 32, T, 0, stream>>>(h_prot, W_dp, ar_dp, er_dp, NP);

  // 3) init segment buffers (emax = enc(-inf), den = 0, o = bias)
  const unsigned NEGINF_ENC = 0x007FFFFFu;
  struct Rel { const int* src; const int* dst; const float* el; const float* er;
               float* emax; float* den; const float* z; float* o; int E; int Ndst; };
  Rel rels[5] = {
    {src_dd, dst_dd, el_dd, er_dd, emax_dd, den_dd, z_dd,   o_dd, E_dd, ND},
    {src_dp, dst_dp, el_dp, er_dp, emax_dp, den_dp, z_dp_d, o_dp, E_dp, NP},
    {src_pd, dst_pd, el_pd, er_pd, emax_pd, den_pd, z_pd_p, o_pd, E_pd, ND},
    {src_pp, dst_pp, el_pp, er_pp, emax_pp, den_pp, z_pp,   o_pp, E_pp, NP},
    {src_sd, dst_sd, el_sd, er_sd, emax_sd, den_sd, z_sd_s, o_sd, E_sd, ND},
  };
  for (int i = 0; i < 5; ++i) {
    const int nh = rels[i].Ndst * 8;
    fill_u32<<<blk(nh), T, 0, stream>>>((unsigned*)rels[i].emax, NEGINF_ENC, nh);
    fill_u32<<<blk(nh), T, 0, stream>>>((unsigned*)rels[i].den, 0u, nh);
  }
  init_bias<<<blk((long long)ND * 128), T, 0, stream>>>(o_dd, b_dd, ND * 128);
  init_bias<<<blk((long long)ND * 128), T, 0, stream>>>(o_pd, b_pd, ND * 128);
  init_bias<<<blk((long long)ND * 128), T, 0, stream>>>(o_sd, b_sd, ND * 128);
  init_bias<<<blk((long long)NP * 128), T, 0, stream>>>(o_dp, b_dp, NP * 128);
  init_bias<<<blk((long long)NP * 128), T, 0, stream>>>(o_pp, b_pp, NP * 128);

  // 4) edge softmax + scatter
  for (int i = 0; i < 5; ++i) {
    Rel& R = rels[i];
    edge_max<<<blk((long long)R.E * 8), T, 0, stream>>>(R.src, R.dst, R.el, R.er,
                                                        (unsigned*)R.emax, R.E);
    emax_decode<<<blk((long long)R.Ndst * 8), T, 0, stream>>>((unsigned*)R.emax,
                                                              R.emax, R.Ndst * 8);
    edge_sum<<<blk((long long)R.E * 8), T, 0, stream>>>(R.src, R.dst, R.el, R.er,
                                                        R.emax, R.den, R.E);
    edge_scatter<<<blk((long long)R.E * 32), T, 0, stream>>>(R.src, R.dst, R.el,
                                                             R.er, R.emax, R.den,
                                                             R.z, R.o, R.E);
  }

  // 5) semantic attention
  semantic_score<<<(ND + 7) / 8, T, 0, stream>>>(o_dd, W1, b1, W2, b2, s_drug, ND, 3, 0);
  semantic_score<<<(ND + 7) / 8, T, 0, stream>>>(o_pd, W1, b1, W2, b2, s_drug, ND, 3, 1);
  semantic_score<<<(ND + 7) / 8, T, 0, stream>>>(o_sd, W1, b1, W2, b2, s_drug, ND, 3, 2);
  semantic_score<<<(NP + 7) / 8, T, 0, stream>>>(o_dp, W1, b1, W2, b2, s_prot, NP, 2, 0);
  semantic_score<<<(NP + 7) / 8, T, 0, stream>>>(o_pp, W1, b1, W2, b2, s_prot, NP, 2, 1);

  float* out = (float*)d_out;
  semantic_combine<<<blk((long long)ND * 128), T, 0, stream>>>(o_dd, o_pd, o_sd,
                                                               s_drug, out, ND, 3);
  semantic_combine<<<blk((long long)NP * 128), T, 0, stream>>>(o_dp, o_pp, o_dp,
                                                               s_prot,
                                                               out + (size_t)ND * 128,
                                                               NP, 2);
}